// HyperNodeConvolution_53644141527472
// MI455X (gfx1250) — compile-verified
//
#include <hip/hip_runtime.h>
#include <math.h>

typedef __attribute__((ext_vector_type(2))) float v2f;
typedef __attribute__((ext_vector_type(8))) float v8f;

#define T_LEN 128
#define D_CH  32
#define OUT_CH 64
#define EPS 1e-5f

__global__ __launch_bounds__(128) void hyper_node_conv_kernel(
    const float* __restrict__ x,     // [S, T]
    const float* __restrict__ w1,    // [32,1,3]
    const float* __restrict__ b1,    // [32]
    const float* __restrict__ w2,    // [1,32,3]
    const float* __restrict__ b2,    // [1]
    const float* __restrict__ w3,    // [64,1,1]
    const float* __restrict__ b3,    // [64]
    const float* __restrict__ g1, const float* __restrict__ be1,
    const float* __restrict__ m1, const float* __restrict__ v1,
    const float* __restrict__ g2, const float* __restrict__ be2,
    const float* __restrict__ m2, const float* __restrict__ v2,
    float* __restrict__ out)         // [S, 64, T]
{
    __shared__ float xs[T_LEN + 2];           // zero-padded input series
    __shared__ float h1s[D_CH][T_LEN + 2];    // zero-padded stage-1 activations
    __shared__ float h2s[T_LEN];              // stage-2 activations

    const int tid = threadIdx.x;              // 0..127, == time index t
    const long s = blockIdx.x;                // series id (B*E*N)
    const float* xp = x + s * (long)T_LEN;

    // ---- load padded input ----
    xs[tid + 1] = xp[tid];
    if (tid == 0) { xs[0] = 0.0f; xs[T_LEN + 1] = 0.0f; }
    if (tid < D_CH) { h1s[tid][0] = 0.0f; h1s[tid][T_LEN + 1] = 0.0f; }
    __syncthreads();

    // ---- stage 1: conv1(1->32,k3,pad1) + BN1 + ReLU ----
    {
        const float x0 = xs[tid], x1 = xs[tid + 1], x2 = xs[tid + 2];
        #pragma unroll 4
        for (int d = 0; d < D_CH; ++d) {
            float sum = w1[d * 3 + 0] * x0 + w1[d * 3 + 1] * x1 + w1[d * 3 + 2] * x2;
            float inv = g1[d] / sqrtf(v1[d] + EPS);
            float y   = (sum + b1[d]) * inv + (be1[d] - m1[d] * inv);
            h1s[d][tid + 1] = y > 0.0f ? y : 0.0f;
        }
    }
    __syncthreads();

    // ---- stage 2: conv2(32->1,k3,pad1) + BN2 + ReLU ----
    {
        float sum = 0.0f;
        #pragma unroll 4
        for (int d = 0; d < D_CH; ++d) {
            sum += w2[d * 3 + 0] * h1s[d][tid]
                 + w2[d * 3 + 1] * h1s[d][tid + 1]
                 + w2[d * 3 + 2] * h1s[d][tid + 2];
        }
        float inv = g2[0] / sqrtf(v2[0] + EPS);
        float y   = (sum + b2[0]) * inv + (be2[0] - m2[0] * inv);
        h2s[tid]  = y > 0.0f ? y : 0.0f;
    }
    __syncthreads();

    // ---- stage 3: conv3(1->64,k1) as rank-1 WMMA: D = w3 (x) h2 + b3 ----
    // V_WMMA_F32_16X16X4_F32, only K=0 populated (K=1..3 slots zeroed).
    // A VGPR0 lanes 0-15 = A[M=lane, K=0]; B VGPR0 lanes 0-15 = B[K=0, N=lane].
    // C/D: VGPR r, lane L -> M = r + (L>=16 ? 8 : 0), N = L & 15.
    const int lane = tid & 31;
    const int half = lane >> 4;     // 0: lanes 0-15, 1: lanes 16-31
    const int l16  = lane & 15;
    const int wave = tid >> 5;      // 4 waves per block
    float* outp = out + s * (long)(OUT_CH * T_LEN);

    for (int nt = 0; nt < 2; ++nt) {
        const int tbase = (wave * 2 + nt) * 16;     // time tile base
        v2f bmat;
        bmat.x = (half == 0) ? h2s[tbase + l16] : 0.0f;
        bmat.y = 0.0f;
        for (int mt = 0; mt < 4; ++mt) {
            const int obase = mt * 16;              // output-channel tile base
            v2f amat;
            amat.x = (half == 0) ? w3[obase + l16] : 0.0f;
            amat.y = 0.0f;
            v8f cmat;
            #pragma unroll
            for (int r = 0; r < 8; ++r)
                cmat[r] = b3[obase + r + half * 8];
            v8f dmat = __builtin_amdgcn_wmma_f32_16x16x4_f32(
                false, amat, false, bmat, (short)0, cmat, false, false);
            #pragma unroll
            for (int r = 0; r < 8; ++r) {
                const int o = obase + r + half * 8;
                outp[o * T_LEN + tbase + l16] = dmat[r];
            }
        }
    }
}

extern "C" void kernel_launch(void* const* d_in, const int* in_sizes, int n_in,
                              void* d_out, int out_size, void* d_ws, size_t ws_size,
                              hipStream_t stream) {
    const float* x   = (const float*)d_in[0];   // node_features [B,E,N,T,1]
    // d_in[1] = edge_features: unused by the reference
    const float* w1  = (const float*)d_in[2];
    const float* b1  = (const float*)d_in[3];
    const float* w2  = (const float*)d_in[4];
    const float* b2  = (const float*)d_in[5];
    const float* w3  = (const float*)d_in[6];
    const float* b3  = (const float*)d_in[7];
    const float* g1  = (const float*)d_in[8];
    const float* be1 = (const float*)d_in[9];
    const float* m1  = (const float*)d_in[10];
    const float* v1  = (const float*)d_in[11];
    const float* g2  = (const float*)d_in[12];
    const float* be2 = (const float*)d_in[13];
    const float* m2  = (const float*)d_in[14];
    const float* v2  = (const float*)d_in[15];
    float* out = (float*)d_out;

    const int S = in_sizes[0] / T_LEN;          // B*E*N = 8192 series
    hipLaunchKernelGGL(hyper_node_conv_kernel, dim3(S), dim3(128), 0, stream,
                       x, w1, b1, w2, b2, w3, b3,
                       g1, be1, m1, v1, g2, be2, m2, v2, out);
}